// ReverseLossLayer_45586782880235
// MI455X (gfx1250) — compile-verified
//
#include <hip/hip_runtime.h>

typedef float v2f __attribute__((ext_vector_type(2)));
typedef float v4f __attribute__((ext_vector_type(4)));
typedef float v8f __attribute__((ext_vector_type(8)));

#define TILES 1250   // 20000 / 16, exact
#define PAIRS 625    // 1250 / 2, exact

// ---------------------------------------------------------------------------
// Kernel 1: pack source points into WMMA B-operand lane layout, two tiles
// interleaved per "pair" so the hot loop uses one global_load_b128 per pair.
// Tile st, lane l (n = st*16 + (l & 15)):
//   lanes 0-15 : b0 = s_n[0] (K=0),  b1 = s_n[1]    (K=1)
//   lanes 16-31: b0 = s_n[2] (K=2),  b1 = ||s_n||^2 (K=3, pairs with A's 1.0)
// Layout: pk[(st>>1)*128 + 4*l + 2*(st&1) + {0,1}]
// ---------------------------------------------------------------------------
__global__ void pack_src_kernel(const float* __restrict__ src,
                                float* __restrict__ pk) {
    int id = blockIdx.x * blockDim.x + threadIdx.x;
    if (id >= TILES * 32) return;
    int st   = id >> 5;
    int lane = id & 31;
    int n = st * 16 + (lane & 15);
    float s0 = src[3 * n + 0];
    float s1 = src[3 * n + 1];
    float s2 = src[3 * n + 2];
    float b0, b1;
    if (lane < 16) { b0 = s0; b1 = s1; }
    else           { b0 = s2; b1 = s0 * s0 + s1 * s1 + s2 * s2; }
    int pr   = st >> 1;
    int half = st & 1;
    pk[pr * 128 + 4 * lane + 2 * half + 0] = b0;
    pk[pr * 128 + 4 * lane + 2 * half + 1] = b1;
}

// Raw v_min_num_f32 / v_min3_num_f32 (avoid fminf's NaN canonicalization).
__device__ __forceinline__ void vmin(float& acc, float x) {
    asm("v_min_num_f32 %0, %0, %1" : "+v"(acc) : "v"(x));
}
__device__ __forceinline__ void vmin3(float& acc, float x, float y) {
    asm("v_min3_num_f32 %0, %0, %1, %2" : "+v"(acc) : "v"(x), "v"(y));
}

// ---------------------------------------------------------------------------
// Kernel 2: per 16-target tile, min over all sources of (||s||^2 - 2 t.s)
// via V_WMMA_F32_16X16X4_F32, then + ||t||^2 and a block partial sum.
// One block per target tile; 8 waves stride the 625 source-tile PAIRS.
// ---------------------------------------------------------------------------
__global__ void __launch_bounds__(256) nn_min_kernel(
        const float* __restrict__ tar,
        const float* __restrict__ pk,
        float* __restrict__ partial) {
    __shared__ float smin[8 * 16];

    const int tid  = threadIdx.x;
    const int lane = tid & 31;
    // readfirstlane -> SGPR: loop trip count is provably wave-uniform.
    const int wave = __builtin_amdgcn_readfirstlane(tid >> 5);
    const int tbase = blockIdx.x * 16;
    const unsigned hi = (unsigned)lane >> 4;   // 0: lanes 0-15, 1: lanes 16-31
    const int r = tbase + (lane & 15);

    // A operand (16x4, M = lane&15):
    //   VGPR0: K=0 (lanes 0-15) = -2*t0 ; K=2 (lanes 16-31) = -2*t2
    //   VGPR1: K=1 (lanes 0-15) = -2*t1 ; K=3 (lanes 16-31) = 1.0 (bias slot)
    v2f a;
    a.x = -2.0f * tar[3 * r + 2 * hi];
    a.y = hi ? 1.0f : (-2.0f * tar[3 * r + 1]);

    v8f c = {0.0f, 0.0f, 0.0f, 0.0f, 0.0f, 0.0f, 0.0f, 0.0f};

    float mA[8], mB[8];
#pragma unroll
    for (int j = 0; j < 8; ++j) { mA[j] = 3.0e38f; mB[j] = 3.0e38f; }

    const float* p = pk + (size_t)(wave * 128) + 4 * lane;
    const int npair = (PAIRS - wave + 7) >> 3;   // #pairs for this wave

    int i = 0;
    // Unroll x2 over pairs: 2 b128 loads, 4 WMMAs, 16 min3 per 4 tiles.
    for (; i + 2 <= npair; i += 2) {
        v4f q0 = *(const v4f*)(p);
        v4f q1 = *(const v4f*)(p + 1024);        // next pair, offset:4096 B
        v2f b00 = {q0.x, q0.y}, b01 = {q0.z, q0.w};
        v2f b10 = {q1.x, q1.y}, b11 = {q1.z, q1.w};
        v8f d00 = __builtin_amdgcn_wmma_f32_16x16x4_f32(
            false, a, false, b00, (short)0, c, false, false);
        v8f d01 = __builtin_amdgcn_wmma_f32_16x16x4_f32(
            false, a, false, b01, (short)0, c, false, false);
        v8f d10 = __builtin_amdgcn_wmma_f32_16x16x4_f32(
            false, a, false, b10, (short)0, c, false, false);
        v8f d11 = __builtin_amdgcn_wmma_f32_16x16x4_f32(
            false, a, false, b11, (short)0, c, false, false);
#pragma unroll
        for (int j = 0; j < 8; ++j) vmin3(mA[j], d00[j], d01[j]);
#pragma unroll
        for (int j = 0; j < 8; ++j) vmin3(mB[j], d10[j], d11[j]);
        p += 2048;                               // 2 pairs * 8 waves * 128
    }
    if (i < npair) {
        v4f q0 = *(const v4f*)(p);
        v2f b00 = {q0.x, q0.y}, b01 = {q0.z, q0.w};
        v8f d00 = __builtin_amdgcn_wmma_f32_16x16x4_f32(
            false, a, false, b00, (short)0, c, false, false);
        v8f d01 = __builtin_amdgcn_wmma_f32_16x16x4_f32(
            false, a, false, b01, (short)0, c, false, false);
#pragma unroll
        for (int j = 0; j < 8; ++j) vmin3(mA[j], d00[j], d01[j]);
    }
#pragma unroll
    for (int j = 0; j < 8; ++j) vmin(mA[j], mB[j]);

    // Cross-lane min within each 16-lane half (XOR masks 1,2,4,8 never cross
    // bit 4, so the two row-groups of the C/D layout stay separate).
#pragma unroll
    for (int m = 8; m >= 1; m >>= 1) {
#pragma unroll
        for (int j = 0; j < 8; ++j)
            vmin(mA[j], __shfl_xor(mA[j], m, 32));
    }

    // VGPR j / lanes 0-15 -> row j ; lanes 16-31 -> row j+8.
    if (lane == 0 || lane == 16) {
#pragma unroll
        for (int j = 0; j < 8; ++j)
            smin[wave * 16 + j + 8 * hi] = mA[j];
    }
    __syncthreads();

    // Wave 0 combines the 8 waves, adds ||t||^2, produces block partial sum.
    if (tid < 32) {
        float v = 0.0f;
        if (tid < 16) {
            float m = smin[tid];
#pragma unroll
            for (int w = 1; w < 8; ++w) m = fminf(m, smin[w * 16 + tid]);
            int rr = tbase + tid;
            float t0 = tar[3 * rr + 0];
            float t1 = tar[3 * rr + 1];
            float t2 = tar[3 * rr + 2];
            v = m + t0 * t0 + t1 * t1 + t2 * t2;
        }
#pragma unroll
        for (int m = 16; m >= 1; m >>= 1) v += __shfl_xor(v, m, 32);
        if (tid == 0) partial[blockIdx.x] = v;
    }
}

// ---------------------------------------------------------------------------
// Kernel 3: deterministic fixed-topology reduction of 1250 block partials.
// ---------------------------------------------------------------------------
__global__ void __launch_bounds__(256) final_reduce_kernel(
        const float* __restrict__ partial, float* __restrict__ out) {
    __shared__ float s[256];
    int tid = threadIdx.x;
    float acc = 0.0f;
    for (int i = tid; i < TILES; i += 256) acc += partial[i];
    s[tid] = acc;
    __syncthreads();
    for (int w = 128; w >= 1; w >>= 1) {
        if (tid < w) s[tid] += s[tid + w];
        __syncthreads();
    }
    if (tid == 0) out[0] = 0.5f * s[0];
}

// ---------------------------------------------------------------------------
extern "C" void kernel_launch(void* const* d_in, const int* in_sizes, int n_in,
                              void* d_out, int out_size, void* d_ws, size_t ws_size,
                              hipStream_t stream) {
    const float* src = (const float*)d_in[0];   // [20000, 3] f32
    const float* tar = (const float*)d_in[1];   // [20000, 3] f32
    float* out = (float*)d_out;                 // scalar f32

    float* pk      = (float*)d_ws;              // PAIRS*128 floats = 320 KB
    float* partial = pk + PAIRS * 128;          // TILES floats

    pack_src_kernel<<<(TILES * 32 + 255) / 256, 256, 0, stream>>>(src, pk);
    nn_min_kernel<<<TILES, 256, 0, stream>>>(tar, pk, partial);
    final_reduce_kernel<<<1, 256, 0, stream>>>(partial, out);
}